// SuperpositionNeuron_45569603010752
// MI455X (gfx1250) — compile-verified
//
#include <hip/hip_runtime.h>

typedef __attribute__((ext_vector_type(2))) float v2f;
typedef __attribute__((ext_vector_type(4))) float v4f;
typedef __attribute__((ext_vector_type(8))) float v8f;
typedef __attribute__((vector_size(16))) int v4i_vs;
typedef __attribute__((address_space(1))) v4i_vs* g_v4i_ptr;   // global (AS1)
typedef __attribute__((address_space(3))) v4i_vs* l_v4i_ptr;   // LDS (AS3)

#define THREADS 128
#define WAVES 4
#define ROWS_PER_BLOCK 64      // 4 waves * 16 rows
#define DDIM 128               // input dim (K)
#define SDIM 4                 // states
#define XS_ROW_F 132           // 128 + 4 floats pad -> bank-conflict-free A reads
#define XS_WAVE_F (16 * XS_ROW_F)
#define WT_ROW_F 132           // padded transposed W rows

#if defined(__gfx1250__) && __has_builtin(__builtin_amdgcn_global_load_async_to_lds_b128)
#define USE_ASYNC_LDS 1
#else
#define USE_ASYNC_LDS 0
#endif

__device__ __forceinline__ void wait_async_then_barrier() {
#if USE_ASYNC_LDS
#if __has_builtin(__builtin_amdgcn_s_wait_asynccnt)
    __builtin_amdgcn_s_wait_asynccnt(0);
#else
    asm volatile("s_wait_asynccnt 0" ::: "memory");
#endif
#endif
    __syncthreads();
}

__global__ __launch_bounds__(THREADS)
void superposition_kernel(const float* __restrict__ x,
                          const float* __restrict__ W,
                          const float* __restrict__ bvec,
                          const float* __restrict__ Mm,
                          const float* __restrict__ thr,
                          const float* __restrict__ noise,
                          const float* __restrict__ u,
                          const float* __restrict__ gumbel,
                          float* __restrict__ out)
{
    __shared__ alignas(16) float xs[WAVES * XS_WAVE_F];         // 33792 B
    __shared__ alignas(16) float wt[16 * WT_ROW_F];             //  8448 B
    __shared__ alignas(16) float stateT[ROWS_PER_BLOCK * SDIM]; //  1024 B

    const int t    = threadIdx.x;
    const int row0 = blockIdx.x * ROWS_PER_BLOCK;

    // ---- Stage x tile: 64 rows x 128 f32, coalesced 16B per lane.
    //      gfx1250 path: async global->LDS DMA (no VGPR round-trip, ASYNCcnt).
    const v4f* x4 = reinterpret_cast<const v4f*>(x);
    #pragma unroll
    for (int i = 0; i < 16; ++i) {
        int idx = i * THREADS + t;         // 0..2047
        int row = idx >> 5;                // 0..63
        int kq  = idx & 31;                // float4 column
        int wv = row >> 4, r = row & 15;
        const v4f* gp = &x4[(size_t)(row0 + row) * (DDIM / 4) + kq];
        float*     lp = &xs[wv * XS_WAVE_F + r * XS_ROW_F + kq * 4];
#if USE_ASYNC_LDS
        __builtin_amdgcn_global_load_async_to_lds_b128(
            (g_v4i_ptr)(v4i_vs*)(void*)const_cast<v4f*>(gp),
            (l_v4i_ptr)(v4i_vs*)(void*)lp,
            0, 0);
#else
        v4f g = __builtin_nontemporal_load(gp);
        *reinterpret_cast<v4f*>(lp) = g;
#endif
    }

    // ---- Stage W transposed+padded: wt[n*132 + k] = (n<4 && k<128) ? W[k*4+n] : 0
    for (int idx = t; idx < 16 * WT_ROW_F; idx += THREADS) {
        int n = idx / WT_ROW_F;
        int k = idx - n * WT_ROW_F;
        float v = 0.0f;
        if (n < SDIM && k < DDIM) v = W[k * SDIM + n];
        wt[idx] = v;
    }

    wait_async_then_barrier();

    const int wave = t >> 5;
    const int lane = t & 31;
    const int half = lane >> 4;        // 0: K{0,1} ; 1: K{2,3}
    const int n    = lane & 15;        // N column (only 0..3 valid)

    const float* xw = &xs[wave * XS_WAVE_F];

    // ---- GEMM tile: D[16x16] += A[16x4] * B[4x16] over K=128 (32 steps, f32 WMMA)
    v8f c = {};
    #pragma unroll
    for (int k0 = 0; k0 < DDIM; k0 += 4) {
        // A: lane r (=lane&15) holds x[r][k0+2*half .. +1]  (ISA 16x4 f32 A layout)
        v2f a = *reinterpret_cast<const v2f*>(&xw[(lane & 15) * XS_ROW_F + k0 + 2 * half]);
        // B: lane holds W[k0+2*half][n], W[k0+2*half+1][n]   (4x16 f32 B layout)
        v2f b = *reinterpret_cast<const v2f*>(&wt[n * WT_ROW_F + k0 + 2 * half]);
        c = __builtin_amdgcn_wmma_f32_16x16x4_f32(false, a, false, b, (short)0, c,
                                                  false, false);
    }

    // ---- bias + sigmoid, scatter valid columns (N<4) to LDS state tile
    union { v8f v; float f[8]; } cu; cu.v = c;
    float bn = (n < SDIM) ? bvec[n] : 0.0f;
    if (n < SDIM) {
        #pragma unroll
        for (int j = 0; j < 8; ++j) {
            int m = j + 8 * half;                       // row within 16-row tile
            float z = cu.f[j] + bn;
            float s = 1.0f / (1.0f + __expf(-z));
            stateT[(wave * 16 + m) * SDIM + n] = s;
        }
    }
    __syncthreads();

    // ---- Per-row epilogue: interference, collapse decision, argmax sampling
    if (t < ROWS_PER_BLOCK) {
        int gr = row0 + t;
        v4f st = *reinterpret_cast<const v4f*>(&stateT[t * SDIM]);

        float inter[SDIM];
        #pragma unroll
        for (int s = 0; s < SDIM; ++s) {
            inter[s] = st.x * Mm[s * SDIM + 0] + st.y * Mm[s * SDIM + 1]
                     + st.z * Mm[s * SDIM + 2] + st.w * Mm[s * SDIM + 3];
        }

        float prob = 1.0f / (1.0f + __expf(-(noise[gr] + thr[0])));
        bool should = u[gr] < prob;

        v4f gmb = __builtin_nontemporal_load(&reinterpret_cast<const v4f*>(gumbel)[gr]);
        float zb0 = inter[0] + gmb.x, zb1 = inter[1] + gmb.y;
        float zb2 = inter[2] + gmb.z, zb3 = inter[3] + gmb.w;

        // first-max argmax (matches jnp.argmax tie-breaking: strict >)
        float best = zb0;     float meas = inter[0];
        if (zb1 > best) { best = zb1; meas = inter[1]; }
        if (zb2 > best) { best = zb2; meas = inter[2]; }
        if (zb3 > best) { best = zb3; meas = inter[3]; }

        float mean = 0.25f * (inter[0] + inter[1] + inter[2] + inter[3]);
        __builtin_nontemporal_store(should ? meas : mean, &out[gr]);
    }
}

extern "C" void kernel_launch(void* const* d_in, const int* in_sizes, int n_in,
                              void* d_out, int out_size, void* d_ws, size_t ws_size,
                              hipStream_t stream) {
    const float* x      = (const float*)d_in[0];
    const float* W      = (const float*)d_in[1];
    const float* b      = (const float*)d_in[2];
    const float* M      = (const float*)d_in[3];
    const float* thr    = (const float*)d_in[4];
    const float* noise  = (const float*)d_in[5];
    const float* u      = (const float*)d_in[6];
    const float* gumbel = (const float*)d_in[7];
    float* out = (float*)d_out;

    int Brows = in_sizes[0] / DDIM;
    int grid  = (Brows + ROWS_PER_BLOCK - 1) / ROWS_PER_BLOCK;
    superposition_kernel<<<grid, THREADS, 0, stream>>>(x, W, b, M, thr, noise, u,
                                                       gumbel, out);
}